// DotProductPredictor_53626961658087
// MI455X (gfx1250) — compile-verified
//
#include <hip/hip_runtime.h>

typedef float v2f __attribute__((ext_vector_type(2)));
typedef float v8f __attribute__((ext_vector_type(8)));

// One wave32 computes 16 edge dot-products as diag(A(16xK) * B(Kx16)) using
// V_WMMA_F32_16X16X4_F32 accumulated over K=128 in 32 steps.
//
// A layout (16x4 f32): lanes 0-15 -> M=lane, K={0,1}; lanes 16-31 -> M=lane-16, K={2,3}
// B layout (4x16 f32): lanes 0-15 -> N=lane, K={0,1}; lanes 16-31 -> N=lane-16, K={2,3}
// => every lane loads one float2 from "its" src row and one from "its" dst row per step.
__global__ __launch_bounds__(256) void u_dot_v_wmma_kernel(
    const float* __restrict__ hsrc,
    const float* __restrict__ hdst,
    const int*   __restrict__ src,
    const int*   __restrict__ dst,
    float*       __restrict__ out,
    int nE)
{
    const int lane = threadIdx.x & 31;
    const int wave = threadIdx.x >> 5;

    // 8 waves per block, 16 edges per wave -> 128 edges per block
    const int e0 = blockIdx.x * 128 + wave * 16;
    if (e0 >= nE) return;                 // wave-uniform: EXEC stays all-ones below

    const int m  = lane & 15;             // edge slot within the 16-edge tile
    const int e  = e0 + m;
    const int ec = (e < nE) ? e : (nE - 1);   // clamp (keeps EXEC full for WMMA)

    const long srow = (long)src[ec] * 128;
    const long drow = (long)dst[ec] * 128;
    const int  bg   = (lane >> 4) << 1;   // K sub-offset: 0 for lanes 0-15, 2 for 16-31

    const float* ap = hsrc + srow + bg;
    const float* bp = hdst + drow + bg;

    v8f c = {};
    #pragma unroll
    for (int kb = 0; kb < 128; kb += 4) {
        v2f a = *(const v2f*)(ap + kb);   // 8B aligned: kb%4==0, bg even
        v2f b = *(const v2f*)(bp + kb);
        // (neg_a, A, neg_b, B, c_mod, C, reuse_a, reuse_b)
        c = __builtin_amdgcn_wmma_f32_16x16x4_f32(
                false, a, false, b, (short)0, c, false, false);
    }

    // Diagonal extraction from the 16x16 f32 C/D layout:
    //   (m,m) for m=0..7  -> VGPR m,    lane m
    //   (m,m) for m=8..15 -> VGPR m-8,  lane m+16
    const bool writer = (lane < 8) || (lane >= 24);
    const int  idx    = (lane < 8) ? lane : (lane - 24);
    float val = c[0];
    #pragma unroll
    for (int i = 1; i < 8; ++i) val = (idx == i) ? c[i] : val;

    if (writer && e < nE) out[e] = val;
}

extern "C" void kernel_launch(void* const* d_in, const int* in_sizes, int n_in,
                              void* d_out, int out_size, void* d_ws, size_t ws_size,
                              hipStream_t stream) {
    const float* h_user = (const float*)d_in[0];
    const float* h_item = (const float*)d_in[1];
    const int*   src0   = (const int*)d_in[2];
    const int*   dst0   = (const int*)d_in[3];
    const int*   src1   = (const int*)d_in[4];
    const int*   dst1   = (const int*)d_in[5];
    const int*   src2   = (const int*)d_in[6];
    const int*   dst2   = (const int*)d_in[7];
    float* out = (float*)d_out;

    const int E = in_sizes[2];            // edges per canonical type
    const int blocks = (E + 127) / 128;   // 128 edges per 256-thread block

    // etype 0: user -> user
    u_dot_v_wmma_kernel<<<blocks, 256, 0, stream>>>(h_user, h_user, src0, dst0, out,         E);
    // etype 1: user -> item
    u_dot_v_wmma_kernel<<<blocks, 256, 0, stream>>>(h_user, h_item, src1, dst1, out + E,     E);
    // etype 2: item -> user
    u_dot_v_wmma_kernel<<<blocks, 256, 0, stream>>>(h_item, h_user, src2, dst2, out + 2 * E, E);
}